// VectorQuantizer_71047349010729
// MI455X (gfx1250) — compile-verified
//
#include <hip/hip_runtime.h>

typedef __attribute__((ext_vector_type(2))) float v2f;
typedef __attribute__((ext_vector_type(8))) float v8f;

#define N_ROWS 16384
#define DDIM   256
#define KCB    8192
#define MT     4                 // 16-row M tiles per block
#define ROWS_PB (MT * 16)        // 64 rows per block
#define LDS_STRIDE 260           // z pad: conflict-free b64 A reads
#define BSTRIDE 68               // B chunk row pad: conflict-free b64 B reads
#define NTILES_PER_WAVE (KCB / 16 / 8)   // 64

// ---------------------------------------------------------------------------
// async B staging: 16 rows x 64 floats (4 KB) per chunk, 8 x b128 per lane.
// lane (r = lane>>1, s = lane&1); instr i covers floats s*4 + 8*i .. +4
// INST_OFFSET advances global and LDS addresses together (32 B per step).
// ---------------------------------------------------------------------------
__device__ __forceinline__ void issue_chunk(const float* g, unsigned lds) {
  asm volatile(
      "global_load_async_to_lds_b128 %0, %1, off\n\t"
      "global_load_async_to_lds_b128 %0, %1, off offset:32\n\t"
      "global_load_async_to_lds_b128 %0, %1, off offset:64\n\t"
      "global_load_async_to_lds_b128 %0, %1, off offset:96\n\t"
      "global_load_async_to_lds_b128 %0, %1, off offset:128\n\t"
      "global_load_async_to_lds_b128 %0, %1, off offset:160\n\t"
      "global_load_async_to_lds_b128 %0, %1, off offset:192\n\t"
      "global_load_async_to_lds_b128 %0, %1, off offset:224"
      :: "v"(lds), "v"(g) : "memory");
}
#define WAIT_ASYNC_LE(n) asm volatile("s_wait_asynccnt " #n ::: "memory")

// ---------------------------------------------------------------------------
// Kernel 1: e_sq[k] = sum_d codebook[k][d]^2  (one wave per codebook row)
// ---------------------------------------------------------------------------
__global__ __launch_bounds__(256) void vq_esq_kernel(const float* __restrict__ cb,
                                                     float* __restrict__ esq) {
  int wave = threadIdx.x >> 5;
  int lane = threadIdx.x & 31;
  int row  = blockIdx.x * 8 + wave;
  const float* p = cb + (size_t)row * DDIM;
  float s = 0.f;
#pragma unroll
  for (int c = 0; c < DDIM; c += 32) {
    float v = p[c + lane];
    s = fmaf(v, v, s);
  }
#pragma unroll
  for (int m = 16; m > 0; m >>= 1) s += __shfl_xor(s, m, 32);
  if (lane == 0) esq[row] = s;
}

// ---------------------------------------------------------------------------
// Kernel 2: fused GEMM (f32 WMMA) + argmin + gather + partial loss.
// Block = 256 threads (8 waves), 64 z-rows (four 16-row M tiles).
// B is double-buffered per wave in LDS via global_load_async_to_lds_b128.
// ---------------------------------------------------------------------------
__global__ __launch_bounds__(256) void vq_main_kernel(const float* __restrict__ z,
                                                      const float* __restrict__ cb,
                                                      const float* __restrict__ esq,
                                                      float* __restrict__ out,
                                                      float* __restrict__ partials) {
  __shared__ float zs[ROWS_PB * LDS_STRIDE];
  __shared__ float bbuf[8][2][16 * BSTRIDE];
  __shared__ float candS[8][ROWS_PB];
  __shared__ int   candI[8][ROWS_PB];
  __shared__ int   finalIdx[ROWS_PB];
  __shared__ float red[256];

  const int tid     = threadIdx.x;
  const int rowbase = blockIdx.x * ROWS_PB;

  // ---- stage 64 rows of z into LDS (1040B row pitch is 16B aligned)
  const float4* zsrc = (const float4*)(z + (size_t)rowbase * DDIM);
#pragma unroll
  for (int i = tid; i < ROWS_PB * (DDIM / 4); i += 256) {
    int r  = i >> 6;
    int c4 = i & 63;
    float4 v = zsrc[i];
    *(float4*)(&zs[r * LDS_STRIDE + c4 * 4]) = v;
  }
  __syncthreads();

  const int wave = tid >> 5;
  const int lane = tid & 31;
  const int half = lane >> 4;
  const int l16  = lane & 15;

  float bestS[MT][8];
  int   bestI[MT][8];
#pragma unroll
  for (int mt = 0; mt < MT; ++mt)
#pragma unroll
    for (int v = 0; v < 8; ++v) { bestS[mt][v] = 3.402823466e38f; bestI[mt][v] = 0; }

  // A layout for V_WMMA_F32_16X16X4_F32: lane holds M = lane%16, K = v + 2*(lane>=16)
  const float* abase[MT];
#pragma unroll
  for (int mt = 0; mt < MT; ++mt)
    abase[mt] = &zs[(mt * 16 + l16) * LDS_STRIDE + 2 * half];

  // per-lane async staging addresses (low 32 bits of generic ptr == LDS byte addr)
  const unsigned ldsLane = ((unsigned)(lane >> 1) * BSTRIDE + (unsigned)(lane & 1) * 4) * 4u;
  const unsigned l0 = (unsigned)(uintptr_t)(&bbuf[wave][0][0]) + ldsLane;
  const unsigned l1 = (unsigned)(uintptr_t)(&bbuf[wave][1][0]) + ldsLane;

  // per-lane global base: row (lane>>1) of this wave's first tile, sub-col (lane&1)*4
  const float* gb = cb + ((size_t)(wave * 16 + (lane >> 1)) * DDIM) + (lane & 1) * 4;

  issue_chunk(gb, l0);   // prologue: tile 0, chunk 0 -> buf0

  for (int t = 0; t < NTILES_PER_WAVE; ++t) {
    const int ncol = (wave + t * 8) * 16 + l16;

    v8f acc[MT];
#pragma unroll
    for (int mt = 0; mt < MT; ++mt) acc[mt] = {};

#pragma unroll
    for (int kc = 0; kc < 4; ++kc) {
      // issue next chunk (same tile kc+1, or next tile chunk 0 across the epilogue)
      if (kc < 3) {
        issue_chunk(gb + (kc + 1) * 64, ((kc + 1) & 1) ? l1 : l0);
        WAIT_ASYNC_LE(8);          // current chunk (issued earlier) now complete
      } else if (t < NTILES_PER_WAVE - 1) {
        issue_chunk(gb + 8 * 16 * DDIM, l0);   // next tile chunk0 -> buf0 (4 chunks/tile)
        WAIT_ASYNC_LE(8);
      } else {
        WAIT_ASYNC_LE(0);
      }

      // B layout: lane holds N = lane%16, K = v + 2*(lane>=16)
      const float* bp = &bbuf[wave][kc & 1][0] + l16 * BSTRIDE + 2 * half;
#pragma unroll 4
      for (int kk = 0; kk < 64; kk += 4) {
        v2f b = *(const v2f*)(bp + kk);
#pragma unroll
        for (int mt = 0; mt < MT; ++mt) {
          v2f a = *(const v2f*)(abase[mt] + kc * 64 + kk);
          acc[mt] = __builtin_amdgcn_wmma_f32_16x16x4_f32(false, a, false, b,
                                                          (short)0, acc[mt], false, false);
        }
      }
    }
    gb += 8 * 16 * DDIM;   // advance to this wave's next tile

    const float eq = esq[ncol];
#pragma unroll
    for (int mt = 0; mt < MT; ++mt) {
#pragma unroll
      for (int v = 0; v < 8; ++v) {
        float s = fmaf(-2.f, acc[mt][v], eq);   // z_sq per row is argmin-invariant
        if (s < bestS[mt][v]) { bestS[mt][v] = s; bestI[mt][v] = ncol; }
      }
    }
  }

  // ---- butterfly reduce within the 16-lane halves (each half shares rows v+8*half)
#pragma unroll
  for (int mt = 0; mt < MT; ++mt) {
#pragma unroll
    for (int v = 0; v < 8; ++v) {
#pragma unroll
      for (int m = 1; m <= 8; m <<= 1) {
        float os = __shfl_xor(bestS[mt][v], m, 32);
        int   oi = __shfl_xor(bestI[mt][v], m, 32);
        if (os < bestS[mt][v] || (os == bestS[mt][v] && oi < bestI[mt][v])) {
          bestS[mt][v] = os; bestI[mt][v] = oi;
        }
      }
    }
  }
  if (l16 == 0) {
#pragma unroll
    for (int mt = 0; mt < MT; ++mt)
#pragma unroll
      for (int v = 0; v < 8; ++v) {
        int r = mt * 16 + 8 * half + v;        // C layout: VGPR v -> row v + 8*half
        candS[wave][r] = bestS[mt][v];
        candI[wave][r] = bestI[mt][v];
      }
  }
  __syncthreads();

  // ---- cross-wave reduce: one thread per row, tie-break to smallest index
  if (tid < ROWS_PB) {
    float bs = candS[0][tid]; int bi = candI[0][tid];
#pragma unroll
    for (int w = 1; w < 8; ++w) {
      float s = candS[w][tid]; int i2 = candI[w][tid];
      if (s < bs || (s == bs && i2 < bi)) { bs = s; bi = i2; }
    }
    finalIdx[tid] = bi;
    out[(size_t)N_ROWS * DDIM + rowbase + tid] = (float)bi;   // indices as float
  }
  __syncthreads();

  // ---- gather z_q, write z_q_st (== z_q numerically), accumulate squared error
  float lsum = 0.f;
  for (int i = tid; i < ROWS_PB * DDIM; i += 256) {
    int r = i >> 8;
    int c = i & 255;
    float q  = cb[(size_t)finalIdx[r] * DDIM + c];
    float zv = zs[r * LDS_STRIDE + c];
    out[(size_t)(rowbase + r) * DDIM + c] = q;
    float d = q - zv;
    lsum = fmaf(d, d, lsum);
  }
  red[tid] = lsum;
  __syncthreads();
#pragma unroll
  for (int s = 128; s > 0; s >>= 1) {
    if (tid < s) red[tid] += red[tid + s];
    __syncthreads();
  }
  if (tid == 0) partials[blockIdx.x] = red[0];
}

// ---------------------------------------------------------------------------
// Kernel 3: deterministic final reduction of 256 partials -> 3 scalar losses
// ---------------------------------------------------------------------------
__global__ __launch_bounds__(256) void vq_final_kernel(const float* __restrict__ partials,
                                                       float* __restrict__ out) {
  __shared__ float red[256];
  int tid = threadIdx.x;
  red[tid] = partials[tid];
  __syncthreads();
#pragma unroll
  for (int k = 128; k > 0; k >>= 1) {
    if (tid < k) red[tid] += red[tid + k];
    __syncthreads();
  }
  if (tid == 0) {
    float loss = red[0] / (float)(N_ROWS * DDIM);  // codebook_loss == commitment_loss
    float cw   = loss * 0.25f;
    size_t off = (size_t)N_ROWS * DDIM + N_ROWS;
    out[off + 0] = loss;        // codebook_loss
    out[off + 1] = cw;          // commit_w
    out[off + 2] = loss + cw;   // vq_loss
  }
}

// ---------------------------------------------------------------------------
extern "C" void kernel_launch(void* const* d_in, const int* in_sizes, int n_in,
                              void* d_out, int out_size, void* d_ws, size_t ws_size,
                              hipStream_t stream) {
  const float* z  = (const float*)d_in[0];   // [16,1024,256] f32
  const float* cb = (const float*)d_in[1];   // [8192,256] f32
  float* out = (float*)d_out;
  float* esq      = (float*)d_ws;            // 8192 floats
  float* partials = esq + KCB;               // 256 floats

  vq_esq_kernel  <<<KCB / 8,          256, 0, stream>>>(cb, esq);
  vq_main_kernel <<<N_ROWS / ROWS_PB, 256, 0, stream>>>(z, cb, esq, out, partials);
  vq_final_kernel<<<1,                256, 0, stream>>>(partials, out);
}